// CalculatorDipole_2585570312371
// MI455X (gfx1250) — compile-verified
//
#include <hip/hip_runtime.h>
#include <math.h>

// ---- problem constants (from reference) ----
#define NATOMS 20000
#define NPAIRS 1000000
#define NSIDE  15                      // ceil(60/4)
#define KTOT   (NSIDE*NSIDE*NSIDE)     // 3375 k-vectors
#define KPAD   (KTOT + 1)              // pad to even (3376)
#define NCHUNK (KPAD / 2)              // 1688 WMMA K-chunks (4 inner each), 1688 = 4*422

// ---- workspace layout (in floats) ----
#define WS_KV4    0                         // KPAD * float4  (kx,ky,kz,0)
#define WS_BPK    (WS_KV4 + KPAD*4)         // NCHUNK * 32 lanes * float2  (WMMA-ready B)
#define WS_MUSUM  (WS_BPK + NCHUNK*64)      // 3 floats: sum of dipoles
#define WS_TOTAL  (WS_MUSUM + 4)            // ~121540 floats (~486 KB)

#define TWO_PI   6.283185307179586f
#define FOUR_PI  12.566370614359172f
#define ALPHA_C  0.7071067811865476f        // 1/(sqrt(2)*SMEARING)
#define GAUSS_C  0.7978845608028654f        // 2*alpha/sqrt(pi)
#define SELF_C   0.5319230405352436f        // (2/3)*sqrt(2/pi)/SMEARING^3

typedef __attribute__((ext_vector_type(2))) float v2f;
typedef __attribute__((ext_vector_type(8))) float v8f;

// Branch-free sincos for |x| <~ 1e3 (our phases are <= ~131 rad).
// Cody-Waite 2-term FMA reduction by pi/2 + Cephes minimax polys on |r|<=pi/4.
// Reduction residual ~q*1e-15 << f32 ulp, so accuracy matches libm here,
// without the Payne-Hanek slow path / divergent branch inside the WMMA loop.
__device__ __forceinline__ void fast_sincosf(float x, float* sp, float* cp) {
    const float TWO_OVER_PI = 0.63661977236758134f;
    float q = rintf(x * TWO_OVER_PI);
    int iq = (int)q;
    float r = fmaf(q, -1.5707963705062866e+0f, x);   // -pi/2 (hi, f32-rounded)
    r       = fmaf(q,  4.3711388286737929e-8f, r);   // -(pi/2 - hi)
    float z = r * r;
    float ps = fmaf(z, fmaf(z, -1.9515295891e-4f, 8.3321608736e-3f), -1.6666654611e-1f);
    float sr = fmaf(r * z, ps, r);                                  // sin(r)
    float pc = fmaf(z, fmaf(z, 2.4433157118e-5f, -1.3887316255e-3f), 4.1666645683e-2f);
    float cr = fmaf(z * z, pc, fmaf(z, -0.5f, 1.0f));               // cos(r)
    bool swap = (iq & 1) != 0;
    float ss = swap ? cr : sr;
    float cc = swap ? sr : cr;
    ss = (iq & 2)       ? -ss : ss;   // sin negative for q mod 4 in {2,3}
    cc = ((iq + 1) & 2) ? -cc : cc;   // cos negative for q mod 4 in {1,2}
    *sp = ss; *cp = cc;
}

// ---------------- zero out + workspace ----------------
__global__ void zero_kernel(float* __restrict__ out, float* __restrict__ ws) {
    int t = blockIdx.x * blockDim.x + threadIdx.x;
    if (t < 3*NATOMS) out[t] = 0.0f;
    if (t < WS_TOTAL) ws[t]  = 0.0f;
}

// ---------------- k-vector table ----------------
__global__ void kprep_kernel(const float* __restrict__ cell, float* __restrict__ ws) {
    int k = blockIdx.x * blockDim.x + threadIdx.x;
    if (k >= KPAD) return;
    float kx = 0.f, ky = 0.f, kz = 0.f;
    if (k < KTOT) {
        int iz = k % NSIDE;
        int t  = k / NSIDE;
        int iy = t % NSIDE;
        int ix = t / NSIDE;
        // np.fft.fftfreq(15)*15 ordering: 0..7, -7..-1
        float nx = (float)(ix <= NSIDE/2 ? ix : ix - NSIDE);
        float ny = (float)(iy <= NSIDE/2 ? iy : iy - NSIDE);
        float nz = (float)(iz <= NSIDE/2 ? iz : iz - NSIDE);
        kx = nx * (TWO_PI / cell[0]);   // diagonal cell (BOX * I)
        ky = ny * (TWO_PI / cell[4]);
        kz = nz * (TWO_PI / cell[8]);
    }
    ws[WS_KV4 + 4*k + 0] = kx;
    ws[WS_KV4 + 4*k + 1] = ky;
    ws[WS_KV4 + 4*k + 2] = kz;
    ws[WS_KV4 + 4*k + 3] = 0.f;
}

// ---------------- sum of dipoles (surface term) ----------------
__global__ void dsum_kernel(const float* __restrict__ dip, float* __restrict__ ws) {
    __shared__ float sx[256], sy[256], sz[256];
    int t = threadIdx.x;
    float ax = 0.f, ay = 0.f, az = 0.f;
    for (int i = t; i < NATOMS; i += 256) {
        ax += dip[3*i+0]; ay += dip[3*i+1]; az += dip[3*i+2];
    }
    sx[t] = ax; sy[t] = ay; sz[t] = az;
    __syncthreads();
    for (int off = 128; off > 0; off >>= 1) {
        if (t < off) { sx[t] += sx[t+off]; sy[t] += sy[t+off]; sz[t] += sz[t+off]; }
        __syncthreads();
    }
    if (t == 0) {
        ws[WS_MUSUM + 0] = sx[0];
        ws[WS_MUSUM + 1] = sy[0];
        ws[WS_MUSUM + 2] = sz[0];
    }
}

// ---------------- real-space pair kernel ----------------
__global__ void rspace_kernel(const float* __restrict__ dip,
                              const int*   __restrict__ nidx,
                              const float* __restrict__ nvec,
                              float* __restrict__ out) {
    int p = blockIdx.x * blockDim.x + threadIdx.x;
    if (p >= NPAIRS) return;
    int i = nidx[2*p+0];
    int j = nidx[2*p+1];
    float vx = nvec[3*p+0], vy = nvec[3*p+1], vz = nvec[3*p+2];
    float r2 = vx*vx + vy*vy + vz*vz;
    float r  = sqrtf(r2);
    float ar = ALPHA_C * r;
    float ec = erfcf(ar);
    float ga = GAUSS_C * expf(-ar*ar);
    float ir2 = 1.0f / r2;
    float ir3 = ir2 / r;
    // B = erfc/r^3 + gauss/r^2 ; C = 3 erfc/r^5 + gauss*(2a^2/r^2 + 3/r^4), 2a^2 = 1
    float Bf = ec*ir3 + ga*ir2;
    float Cf = 3.0f*ec*ir3*ir2 + ga*(ir2 + 3.0f*ir2*ir2);

    float mjx = dip[3*j+0], mjy = dip[3*j+1], mjz = dip[3*j+2];
    float mix = dip[3*i+0], miy = dip[3*i+1], miz = dip[3*i+2];
    float dj = vx*mjx + vy*mjy + vz*mjz;
    float di = vx*mix + vy*miy + vz*miz;

    // T @ mu = B*mu - C*vec*(vec.mu); final *0.5
    atomicAdd(&out[3*i+0], 0.5f*(Bf*mjx - Cf*vx*dj));
    atomicAdd(&out[3*i+1], 0.5f*(Bf*mjy - Cf*vy*dj));
    atomicAdd(&out[3*i+2], 0.5f*(Bf*mjz - Cf*vz*dj));
    atomicAdd(&out[3*j+0], 0.5f*(Bf*mix - Cf*vx*di));
    atomicAdd(&out[3*j+1], 0.5f*(Bf*miy - Cf*vy*di));
    atomicAdd(&out[3*j+2], 0.5f*(Bf*miz - Cf*vz*di));
}

// ---------------- k-space pass 1: structure factors -> packed WMMA B ----------------
__global__ void kpass1_kernel(const float* __restrict__ pos,
                              const float* __restrict__ dip,
                              const float* __restrict__ cell,
                              float* __restrict__ ws) {
    int k = blockIdx.x;                 // 0..KTOT-1
    float kx = ws[WS_KV4 + 4*k + 0];
    float ky = ws[WS_KV4 + 4*k + 1];
    float kz = ws[WS_KV4 + 4*k + 2];
    float sc = 0.f, ss = 0.f;
    for (int i = threadIdx.x; i < NATOMS; i += 256) {
        float ph = kx*pos[3*i+0] + ky*pos[3*i+1] + kz*pos[3*i+2];
        float mk = kx*dip[3*i+0] + ky*dip[3*i+1] + kz*dip[3*i+2];
        float s, c;
        fast_sincosf(ph, &s, &c);
        sc += c * mk;
        ss += s * mk;
    }
    __shared__ float rc[256], rs[256];
    int t = threadIdx.x;
    rc[t] = sc; rs[t] = ss;
    __syncthreads();
    for (int off = 128; off > 0; off >>= 1) {
        if (t < off) { rc[t] += rc[t+off]; rs[t] += rs[t+off]; }
        __syncthreads();
    }
    if (t == 0) {
        float ksq = kx*kx + ky*ky + kz*kz;
        float vol = fabsf(cell[0]*cell[4]*cell[8]);
        float fac = 0.f;
        if (ksq > 0.f) fac = 0.5f * (FOUR_PI * expf(-0.5f*ksq) / ksq) / vol; // SMEARING=1
        float scv = rc[0] * fac;
        float ssv = rs[0] * fac;
        float kvv[3] = {kx, ky, kz};
        int chunk = k >> 1;
        int half  = k & 1;
        float* bp = ws + WS_BPK + chunk*64;   // float2 per lane, 32 lanes
        for (int c = 0; c < 3; ++c) {
            bp[(half*16 + c)*2 + 0] = scv * kvv[c];  // cos row (even inner index)
            bp[(half*16 + c)*2 + 1] = ssv * kvv[c];  // sin row (odd inner index)
        }
        // lanes with col>=3 stay zero from zero_kernel
    }
}

// ---------------- k-space pass 2: (N x 2K) @ (2K x 3) via V_WMMA_F32_16X16X4_F32 ----------------
__global__ void __launch_bounds__(32)
kpass2_wmma_kernel(const float* __restrict__ pos,
                   const float* __restrict__ dip,
                   const float* __restrict__ cell,
                   const float* __restrict__ ws,
                   float* __restrict__ out) {
    const int lane  = threadIdx.x;       // 0..31, wave32
    const int col   = lane & 15;         // output column (pot component when <3)
    const int khalf = lane >> 4;         // low lanes: K=0,1 of chunk; high lanes: K=2,3
    const int abase = blockIdx.x * 16;   // 16 atoms per wave; 1250 waves * 16 = 20000

    const int m = abase + col;           // atom whose trig this lane supplies (A fragment)
    const float px = pos[3*m+0], py = pos[3*m+1], pz = pos[3*m+2];

    const float4* kv  = (const float4*)(ws + WS_KV4);
    const v2f*    bpk = (const v2f*)(ws + WS_BPK);

    v8f acc = {0.f,0.f,0.f,0.f,0.f,0.f,0.f,0.f};

    #pragma unroll 4
    for (int it = 0; it < NCHUNK; ++it) {
        float4 k4 = kv[2*it + khalf];
        v2f b = bpk[it*32 + lane];       // prepacked B fragment (zeros for col>=3)
        float ph = k4.x*px + k4.y*py + k4.z*pz;
        v2f a;
        float s, c;
        fast_sincosf(ph, &s, &c);
        a.x = c;                         // even inner index = cos row
        a.y = s;                         // odd  inner index = sin row
        acc = __builtin_amdgcn_wmma_f32_16x16x4_f32(
                  /*neg_a=*/false, a, /*neg_b=*/false, b,
                  /*c_mod=*/(short)0, acc, /*reuse_a=*/false, /*reuse_b=*/false);
    }

    // D layout: lanes 0-15 hold rows M=v (v=0..7), lanes 16-31 hold rows M=v+8
    if (col < 3) {
        const float vol  = fabsf(cell[0]*cell[4]*cell[8]);
        const float surf = 0.5f * FOUR_PI / (3.0f * vol);
        const float ms   = ws[WS_MUSUM + col];
        const int row0 = khalf * 8;
        #pragma unroll
        for (int v = 0; v < 8; ++v) {
            int mm = abase + row0 + v;
            float extra = surf * ms - 0.5f * SELF_C * dip[3*mm + col];
            out[3*mm + col] += acc[v] + extra;   // rspace atomics already done (stream order)
        }
    }
}

// ---------------- launcher ----------------
extern "C" void kernel_launch(void* const* d_in, const int* in_sizes, int n_in,
                              void* d_out, int out_size, void* d_ws, size_t ws_size,
                              hipStream_t stream) {
    const float* dip  = (const float*)d_in[0];   // (20000,3) f32
    const float* cell = (const float*)d_in[1];   // (3,3) f32
    const float* pos  = (const float*)d_in[2];   // (20000,3) f32
    const int*   nidx = (const int*)  d_in[3];   // (1e6,2) i32
    const float* nvec = (const float*)d_in[4];   // (1e6,3) f32
    float* out = (float*)d_out;                  // (20000,3) f32
    float* ws  = (float*)d_ws;                   // needs ~486 KB

    int zn = (WS_TOTAL > 3*NATOMS) ? WS_TOTAL : 3*NATOMS;
    zero_kernel<<<(zn + 255)/256, 256, 0, stream>>>(out, ws);
    kprep_kernel<<<(KPAD + 255)/256, 256, 0, stream>>>(cell, ws);
    dsum_kernel<<<1, 256, 0, stream>>>(dip, ws);
    rspace_kernel<<<(NPAIRS + 255)/256, 256, 0, stream>>>(dip, nidx, nvec, out);
    kpass1_kernel<<<KTOT, 256, 0, stream>>>(pos, dip, cell, ws);
    kpass2_wmma_kernel<<<NATOMS/16, 32, 0, stream>>>(pos, dip, cell, ws, out);
}